// Cace_42571715838070
// MI455X (gfx1250) — compile-verified
//
#include <hip/hip_runtime.h>
#include <hip/hip_bf16.h>

// ---------------------------------------------------------------------------
// CACE edge->node tensor kernel for gfx1250 (MI455X), wave32.
//   A[n,s,a,c] = sum_{edges e: rcv(e)=n} R2[e, L(a), s] * ang[e,a] * enc[e,c]
//   R2[e,l,s]  = sum_r radial[e,r] * W[l][r][s]          (<- v_wmma_f32_16x16x4_f32)
//   out[n,s,b,c] = symmetrizer(A[n,s,:,c])               (8 invariant features)
// ---------------------------------------------------------------------------

typedef __attribute__((ext_vector_type(2))) float v2f;
typedef __attribute__((ext_vector_type(8))) float v8f;

#define N_RBF 8
#define N_ANG 20
#define N_CH 9
#define CUTOFF 5.5f
#define R2_PAD 36  // 36-float rows: keeps s*4 offsets 16B aligned, reduces bank conflicts

__device__ __forceinline__ void atomAddF32(float* p, float v) {
    // relaxed, agent scope, result unused -> no-return global_atomic_add_f32
    __hip_atomic_fetch_add(p, v, __ATOMIC_RELAXED, __HIP_MEMORY_SCOPE_AGENT);
}

__device__ __forceinline__ int zmap(int z) {
    // ZS = (1, 6, 7, 8)
    return z == 1 ? 0 : (z == 6 ? 1 : (z == 7 ? 2 : 3));
}

// ------------------------------- zero scratch ------------------------------
__global__ void cace_zero_kernel(float4* p, size_t n4) {
    size_t i = (size_t)blockIdx.x * blockDim.x + threadIdx.x;
    size_t stride = (size_t)gridDim.x * blockDim.x;
    const float4 z = {0.0f, 0.0f, 0.0f, 0.0f};
    for (; i < n4; i += stride) p[i] = z;
}

// ------------------------------- edge kernel -------------------------------
// 256 threads = 8 waves, each wave owns 32 edges.
__global__ void __launch_bounds__(256)
cace_edge_kernel(const float* __restrict__ pos,
                 const float* __restrict__ shifts,
                 const float* __restrict__ embW,    // [4][3]
                 const float* __restrict__ radW,    // [4][8][8]  (l, r, s)
                 const int* __restrict__ atomZ,
                 const int* __restrict__ eidx,      // [2][E]
                 float* __restrict__ Abuf,          // [N][8][20][9]
                 int E) {
    __shared__ float radLDS[8][32][N_RBF];   // per-wave radial rows
    __shared__ float r2LDS[8][32][R2_PAD];   // per-wave R2: [edge][s*4 + l]

    const int lane = threadIdx.x & 31;
    const int wv = threadIdx.x >> 5;
    const int hl = lane >> 4;        // half-wave select
    const int ln = lane & 15;

    const int e = blockIdx.x * 256 + wv * 32 + lane;
    const bool valid = (e < E);
    const int ei = valid ? e : 0;

    const int snd = eidx[ei];
    const int rcv = eidx[E + ei];

    // --- species embeddings -> channel outer product enc[9] ---
    const int zs = zmap(atomZ[snd]);
    const int zr = zmap(atomZ[rcv]);
    float es[3], er[3];
#pragma unroll
    for (int k = 0; k < 3; ++k) { es[k] = embW[zs * 3 + k]; er[k] = embW[zr * 3 + k]; }
    float enc[N_CH];
#pragma unroll
    for (int i = 0; i < 3; ++i)
#pragma unroll
        for (int j = 0; j < 3; ++j) enc[i * 3 + j] = es[i] * er[j];

    // --- geometry ---
    const float vx = pos[rcv * 3 + 0] - pos[snd * 3 + 0] + shifts[ei * 3 + 0];
    const float vy = pos[rcv * 3 + 1] - pos[snd * 3 + 1] + shifts[ei * 3 + 1];
    const float vz = pos[rcv * 3 + 2] - pos[snd * 3 + 2] + shifts[ei * 3 + 2];
    float len = sqrtf(vx * vx + vy * vy + vz * vz);
    len = fmaxf(len, 1e-9f);
    const float inv = 1.0f / len;
    const float ux = vx * inv, uy = vy * inv, uz = vz * inv;

    // --- Bessel RBF * polynomial cutoff (p = 6) ---
    const float x = len * (1.0f / CUTOFF);
    const float x3 = x * x * x;
    const float x6 = x3 * x3;
    const float x7 = x6 * x;
    const float x8 = x7 * x;
    const float fc = (x < 1.0f) ? (1.0f - 28.0f * x6 + 48.0f * x7 - 21.0f * x8) : 0.0f;
    const float pref = sqrtf(2.0f / CUTOFF) * inv * fc;
    const float w0 = 3.14159265358979323846f / CUTOFF * len;
#pragma unroll
    for (int r = 0; r < N_RBF; ++r)
        radLDS[wv][lane][r] = pref * __sinf((float)(r + 1) * w0);

    __syncthreads();

    // --- per-l radial 8->8 transform via V_WMMA_F32_16X16X4_F32 ------------
    // N columns pack (l = 2g + N/8, s = N%8); K covers rbf index r in slices of 4.
    // A layout: lane L, vgpr v -> M = L%16, K = 2*(L/16)+v
    // B layout: lane L, vgpr v -> K = 2*(L/16)+v, N = L%16
    // D layout: lane L, vgpr j -> M = j + 8*(L/16), N = L%16
    v2f bfrag[2][2];
#pragma unroll
    for (int g = 0; g < 2; ++g)
#pragma unroll
        for (int k = 0; k < 2; ++k) {
            const int l = 2 * g + (ln >> 3);
            const int s = ln & 7;
            const int kb = 4 * k + 2 * hl;
            v2f b;
            b.x = radW[l * 64 + (kb + 0) * 8 + s];
            b.y = radW[l * 64 + (kb + 1) * 8 + s];
            bfrag[g][k] = b;
        }

    // D-store column permutation: store R2 as [edge][s*4 + l] so the scatter
    // can fetch all four l values for one s with a single ds_load_b128.
    const int colp = (ln & 7) * 4 + (ln >> 3);  // + 2*g added below
#pragma unroll
    for (int t = 0; t < 2; ++t) {
        const int me = t * 16 + ln;  // edge row within this wave's 32
        v2f afrag[2];
#pragma unroll
        for (int k = 0; k < 2; ++k) {
            v2f a;
            a.x = radLDS[wv][me][4 * k + 2 * hl + 0];
            a.y = radLDS[wv][me][4 * k + 2 * hl + 1];
            afrag[k] = a;
        }
#pragma unroll
        for (int g = 0; g < 2; ++g) {
            v8f acc = {};
            acc = __builtin_amdgcn_wmma_f32_16x16x4_f32(
                false, afrag[0], false, bfrag[g][0], (short)0, acc, false, false);
            acc = __builtin_amdgcn_wmma_f32_16x16x4_f32(
                false, afrag[1], false, bfrag[g][1], (short)0, acc, false, false);
#pragma unroll
            for (int j = 0; j < 8; ++j)
                r2LDS[wv][t * 16 + j + 8 * hl][colp + 2 * g] = acc[j];
        }
    }

    __syncthreads();

    // --- angular monomials + atomic scatter --------------------------------
    if (valid) {
        const float px[4] = {1.0f, ux, ux * ux, ux * ux * ux};
        const float py[4] = {1.0f, uy, uy * uy, uy * uy * uy};
        const float pz[4] = {1.0f, uz, uz * uz, uz * uz * uz};
        const int LXc[N_ANG] = {0,1,0,0,2,1,1,0,0,0,3,2,2,1,1,1,0,0,0,0};
        const int LYc[N_ANG] = {0,0,1,0,0,1,0,2,1,0,0,1,0,2,1,0,3,2,1,0};
        const int LZc[N_ANG] = {0,0,0,1,0,0,1,0,1,2,0,0,1,0,1,2,0,1,2,3};
        const int LOF[N_ANG] = {0,1,1,1,2,2,2,2,2,2,3,3,3,3,3,3,3,3,3,3};

        float ang[N_ANG];
#pragma unroll
        for (int a = 0; a < N_ANG; ++a)
            ang[a] = px[LXc[a]] * py[LYc[a]] * pz[LZc[a]];

        float* nodeBase = Abuf + (size_t)rcv * (8 * N_ANG * N_CH);
#pragma unroll 1
        for (int s = 0; s < 8; ++s) {
            const float4 rl4 = *(const float4*)&r2LDS[wv][lane][s * 4];
            const float rl[4] = {rl4.x, rl4.y, rl4.z, rl4.w};  // rl[l] = R2[e,l,s]
            float* pbase = nodeBase + s * (N_ANG * N_CH);
#pragma unroll
            for (int a = 0; a < N_ANG; ++a) {
                const float v = rl[LOF[a]] * ang[a];
                float* p = pbase + a * N_CH;
#pragma unroll
                for (int c = 0; c < N_CH; ++c) atomAddF32(p + c, v * enc[c]);
            }
        }
    }
}

// ------------------------------- node kernel -------------------------------
__device__ __forceinline__ constexpr int midx(int l, int lx, int ly) {
    const int base = (l == 0) ? 0 : (l == 1) ? 1 : (l == 2) ? 4 : 10;
    const int d = l - lx;
    return base + d * (d + 1) / 2 + (d - ly);
}
__device__ __forceinline__ constexpr float mnom(int l, int lx, int ly, int lz) {
    const float f[4] = {1.0f, 1.0f, 2.0f, 6.0f};
    return f[l] / (f[lx] * f[ly] * f[lz]);
}

template <int E12, int E13, int E23>
__device__ __forceinline__ float nu3(const float* Av) {
    float acc = 0.0f;
#pragma unroll
    for (int ax = E12; ax >= 0; --ax)
#pragma unroll
        for (int ay = E12 - ax; ay >= 0; --ay) {
            const int az = E12 - ax - ay;
#pragma unroll
            for (int bx = E13; bx >= 0; --bx)
#pragma unroll
                for (int by = E13 - bx; by >= 0; --by) {
                    const int bz = E13 - bx - by;
#pragma unroll
                    for (int cx = E23; cx >= 0; --cx)
#pragma unroll
                        for (int cy = E23 - cx; cy >= 0; --cy) {
                            const int cz = E23 - cx - cy;
                            const float pref = mnom(E12, ax, ay, az) *
                                               mnom(E13, bx, by, bz) *
                                               mnom(E23, cx, cy, cz);
                            acc += pref * Av[midx(E12 + E13, ax + bx, ay + by)] *
                                          Av[midx(E12 + E23, ax + cx, ay + cy)] *
                                          Av[midx(E13 + E23, bx + cx, by + cy)];
                        }
                }
        }
    return acc;
}

__global__ void __launch_bounds__(256)
cace_node_kernel(const float* __restrict__ Abuf, float* __restrict__ out, int N) {
    const int tid = blockIdx.x * blockDim.x + threadIdx.x;
    const int total = N * 8 * N_CH;
    if (tid >= total) return;
    const int c = tid % N_CH;
    const int s = (tid / N_CH) & 7;
    const int n = tid / (N_CH * 8);

    float Av[N_ANG];
    const float* ap = Abuf + ((size_t)n * 8 + s) * (N_ANG * N_CH) + c;
#pragma unroll
    for (int a = 0; a < N_ANG; ++a) Av[a] = ap[a * N_CH];

    float f[8];
    f[0] = Av[0];
    // nu=2, l=1..3 (multinomial-weighted sums of squares)
    f[1] = Av[1] * Av[1] + Av[2] * Av[2] + Av[3] * Av[3];
    f[2] = Av[4] * Av[4] + 2.f * Av[5] * Av[5] + 2.f * Av[6] * Av[6] +
           Av[7] * Av[7] + 2.f * Av[8] * Av[8] + Av[9] * Av[9];
    f[3] = Av[10] * Av[10] + 3.f * Av[11] * Av[11] + 3.f * Av[12] * Av[12] +
           3.f * Av[13] * Av[13] + 6.f * Av[14] * Av[14] + 3.f * Av[15] * Av[15] +
           Av[16] * Av[16] + 3.f * Av[17] * Av[17] + 3.f * Av[18] * Av[18] +
           Av[19] * Av[19];
    // nu=3 combos in reference loop order
    f[4] = nu3<1, 1, 1>(Av);
    f[5] = nu3<1, 1, 2>(Av);
    f[6] = nu3<1, 2, 1>(Av);
    f[7] = nu3<2, 1, 1>(Av);

    float* op = out + ((size_t)n * 8 + s) * (8 * N_CH) + c;
#pragma unroll
    for (int b = 0; b < 8; ++b) op[b * N_CH] = f[b];
}

// ------------------------------- launcher ----------------------------------
extern "C" void kernel_launch(void* const* d_in, const int* in_sizes, int n_in,
                              void* d_out, int out_size, void* d_ws, size_t ws_size,
                              hipStream_t stream) {
    const float* pos    = (const float*)d_in[0];
    const float* shifts = (const float*)d_in[1];
    const float* embW   = (const float*)d_in[2];
    const float* radW   = (const float*)d_in[3];
    const int*   atomZ  = (const int*)d_in[4];
    const int*   eidx   = (const int*)d_in[5];

    const int N = in_sizes[0] / 3;
    const int E = in_sizes[5] / 2;

    float* Abuf = (float*)d_ws;                       // [N][8][20][9]
    const size_t Aelems = (size_t)N * 8 * N_ANG * N_CH;

    cace_zero_kernel<<<1024, 256, 0, stream>>>((float4*)Abuf, Aelems / 4);

    const int eblocks = (E + 255) / 256;
    cace_edge_kernel<<<eblocks, 256, 0, stream>>>(pos, shifts, embW, radW, atomZ,
                                                  eidx, Abuf, E);

    const int nthreads = N * 8 * N_CH;
    cace_node_kernel<<<(nthreads + 255) / 256, 256, 0, stream>>>(Abuf, (float*)d_out, N);
}